// ExampleModel_15238543966682
// MI455X (gfx1250) — compile-verified
//
#include <hip/hip_runtime.h>
#include <hip/hip_bf16.h>

#define N_NODES   100000
#define N_EDGES   1200000
#define N_GRAPHS  128
#define IN_CH     30
#define HID       64

typedef float v2f __attribute__((ext_vector_type(2)));
typedef float v8f __attribute__((ext_vector_type(8)));

#if defined(__has_builtin)
#if __has_builtin(__builtin_amdgcn_global_load_async_to_lds_b32)
#define HAVE_ASYNC_LDS 1
typedef __attribute__((address_space(1))) int gas_int;   // global (prints as __device__ int*)
typedef __attribute__((address_space(3))) int las_int;   // LDS    (prints as __shared__ int*)
#endif
#endif

// ---------------------------------------------------------------- zero ----
__global__ void zero_f32(float* __restrict__ p, int n) {
    int i = blockIdx.x * blockDim.x + threadIdx.x;
    if (i < n) p[i] = 0.0f;
}

// --------------------------------------------------- edge scatter-add ----
// agg[dst][c] += feat[src][c]; agg has fixed stride HID(=64).
// One thread per (edge, channel); channels padded to power of two (1<<logCPE).
__global__ void scatter_edges(const float* __restrict__ feat, int feat_stride,
                              int C, int logCPE,
                              const int* __restrict__ src, const int* __restrict__ dst,
                              float* __restrict__ agg, long total) {
    long t = (long)blockIdx.x * blockDim.x + threadIdx.x;
    if (t >= total) return;
    int c = (int)(t & ((1 << logCPE) - 1));
    if (c >= C) return;
    int e = (int)(t >> logCPE);
    int s = src[e], d = dst[e];
    atomicAdd(&agg[(long)d * HID + c], feat[(long)s * feat_stride + c]);
}

// --------------------------------------------------- fused GIN MLP -------
// h_out = relu(((1+eps)*x + agg) @ w1 + b1) @ w2 + b2, via fp32 WMMA 16x16x4.
// Block = 128 threads (4 waves); each wave owns one 16-node tile.
//
// Weights staged in fragment-packed LDS layout: for k-group g = 2*kk4 + hi,
//   wp[g*RS + 2n + p] = w[4*kk4 + 2*hi + p][n]   (p = 0,1)
// so a B fragment is ONE aligned ds_load_b64 into the WMMA source pair.
// RS=160 (RS%64==32): the two wave halves (g, g+1) hit disjoint bank sets.
// z/y tiles use stride ZS=68: A-frag banks are 4m+k -> conflict-free.
#define RS 160
#define ZS 68
#define NGROUPS 32          // Kpad<=64 -> up to 32 (kk4,hi) groups
#define TILES_PER_BLOCK 4

__device__ __forceinline__ int pack_slot(int k, int n) {
    int g = (k >> 2) * 2 + ((k >> 1) & 1);
    return g * RS + 2 * n + (k & 1);
}

__global__ __launch_bounds__(128)
void gin_mlp(const float* __restrict__ xin, int in_stride, int Kin, int Kpad, int logKpad,
             const float* __restrict__ agg,
             const float* __restrict__ eps_ptr, int eps_idx,
             const float* __restrict__ w1, const float* __restrict__ b1,
             const float* __restrict__ w2, const float* __restrict__ b2,
             float* __restrict__ hout, int num_tiles)
{
    __shared__ float w1p[NGROUPS * RS];
    __shared__ float w2p[NGROUPS * RS];
    __shared__ float b1s[HID], b2s[HID];
    __shared__ float zs[TILES_PER_BLOCK][16 * ZS];

    const int tid = threadIdx.x;

    // ---- stage weights into packed LDS layout ----
#ifdef HAVE_ASYNC_LDS
    // async global->LDS with per-lane scattered LDS destinations (ASYNCcnt)
    for (int idx = tid; idx < Kin * HID; idx += 128) {
        int k = idx >> 6, n = idx & 63;
        __builtin_amdgcn_global_load_async_to_lds_b32(
            (gas_int*)(w1 + idx),
            (las_int*)&w1p[pack_slot(k, n)], 0, 0);
    }
    for (int idx = tid; idx < HID * HID; idx += 128) {
        int k = idx >> 6, n = idx & 63;
        __builtin_amdgcn_global_load_async_to_lds_b32(
            (gas_int*)(w2 + idx),
            (las_int*)&w2p[pack_slot(k, n)], 0, 0);
    }
#else
    for (int idx = tid; idx < Kin * HID; idx += 128) {
        int k = idx >> 6, n = idx & 63;
        w1p[pack_slot(k, n)] = w1[idx];
    }
    for (int idx = tid; idx < HID * HID; idx += 128) {
        int k = idx >> 6, n = idx & 63;
        w2p[pack_slot(k, n)] = w2[idx];
    }
#endif
    // zero-pad w1 rows [Kin,Kpad) (layer 0: K 30->32)
    for (int idx = Kin * HID + tid; idx < Kpad * HID; idx += 128) {
        int k = idx >> 6, n = idx & 63;
        w1p[pack_slot(k, n)] = 0.0f;
    }
    if (tid < HID) { b1s[tid] = b1[tid]; b2s[tid] = b2[tid]; }

#ifdef HAVE_ASYNC_LDS
    asm volatile("s_wait_asynccnt 0" ::: "memory");
#endif
    __syncthreads();   // only block-wide barrier; reached by all threads

    const float scale = 1.0f + eps_ptr[eps_idx];
    const int wave = tid >> 5;
    const int lane = tid & 31;
    const int ln   = lane & 15;      // column / row-in-half index
    const int hi   = lane >> 4;      // which half of the wave

    const int tile = blockIdx.x * TILES_PER_BLOCK + wave;
    if (tile >= num_tiles) return;   // wave-uniform; EXEC stays full for WMMA
    const long node0 = (long)tile * 16;
    float* z = zs[wave];

    // Build z tile: z[m][c] = (1+eps)*x + agg  (zero-pad c in [Kin,Kpad))
    for (int idx = lane; idx < (16 << logKpad); idx += 32) {
        int m = idx >> logKpad;
        int c = idx & (Kpad - 1);
        float v = 0.0f;
        if (c < Kin)
            v = scale * xin[(node0 + m) * in_stride + c] + agg[(node0 + m) * HID + c];
        z[m * ZS + c] = v;
    }
    asm volatile("s_wait_dscnt 0" ::: "memory");  // cross-lane LDS visibility

    // ---- GEMM1: [16 x Kpad] @ [Kpad x 64] ----
    v8f acc[4];
    #pragma unroll
    for (int nt = 0; nt < 4; ++nt) acc[nt] = (v8f){};
    const int nk4 = Kpad >> 2;
    for (int kk4 = 0; kk4 < nk4; ++kk4) {
        v2f a = *(const v2f*)&z[ln * ZS + kk4 * 4 + 2 * hi];
        const float* bp = &w1p[(kk4 * 2 + hi) * RS + 2 * ln];
        #pragma unroll
        for (int nt = 0; nt < 4; ++nt) {
            v2f b = *(const v2f*)&bp[32 * nt];
            acc[nt] = __builtin_amdgcn_wmma_f32_16x16x4_f32(
                false, a, false, b, (short)0, acc[nt], false, false);
        }
    }

    // bias + ReLU, stage y back into this wave's z slot (z is dead now)
    #pragma unroll
    for (int nt = 0; nt < 4; ++nt) {
        const float bb = b1s[nt * 16 + ln];
        #pragma unroll
        for (int r = 0; r < 8; ++r) {
            float v = acc[nt][r] + bb;
            z[(8 * hi + r) * ZS + nt * 16 + ln] = v > 0.0f ? v : 0.0f;
        }
    }
    asm volatile("s_wait_dscnt 0" ::: "memory");

    // ---- GEMM2: [16 x 64] @ [64 x 64] ----
    #pragma unroll
    for (int nt = 0; nt < 4; ++nt) acc[nt] = (v8f){};
    for (int kk4 = 0; kk4 < (HID >> 2); ++kk4) {
        v2f a = *(const v2f*)&z[ln * ZS + kk4 * 4 + 2 * hi];
        const float* bp = &w2p[(kk4 * 2 + hi) * RS + 2 * ln];
        #pragma unroll
        for (int nt = 0; nt < 4; ++nt) {
            v2f b = *(const v2f*)&bp[32 * nt];
            acc[nt] = __builtin_amdgcn_wmma_f32_16x16x4_f32(
                false, a, false, b, (short)0, acc[nt], false, false);
        }
    }

    // bias + store h_out
    #pragma unroll
    for (int nt = 0; nt < 4; ++nt) {
        const float bb = b2s[nt * 16 + ln];
        #pragma unroll
        for (int r = 0; r < 8; ++r) {
            hout[(node0 + 8 * hi + r) * HID + nt * 16 + ln] = acc[nt][r] + bb;
        }
    }
}

// --------------------------------------------------- graph pooling -------
__global__ void pool_nodes(const float* __restrict__ h, const int* __restrict__ batch,
                           float* __restrict__ pooled, long total) {
    long t = (long)blockIdx.x * blockDim.x + threadIdx.x;
    if (t >= total) return;
    int c = (int)(t & 63);
    int node = (int)(t >> 6);
    atomicAdd(&pooled[(long)batch[node] * HID + c], h[(long)node * HID + c]);
}

__global__ void predict(const float* __restrict__ pooled, const float* __restrict__ w_out,
                        const float* __restrict__ b_out, float* __restrict__ pred) {
    __shared__ float red[HID];
    int g = blockIdx.x, c = threadIdx.x;
    red[c] = pooled[(long)g * HID + c] * w_out[c];
    __syncthreads();
    for (int s = 32; s > 0; s >>= 1) {
        if (c < s) red[c] += red[c + s];
        __syncthreads();
    }
    if (c == 0) pred[g] = red[0] + b_out[0];
}

// ------------------------------------------------------------ launch -----
extern "C" void kernel_launch(void* const* d_in, const int* in_sizes, int n_in,
                              void* d_out, int out_size, void* d_ws, size_t ws_size,
                              hipStream_t stream) {
    const float* x     = (const float*)d_in[0];
    const int*   eidx  = (const int*)  d_in[1];
    const int*   batch = (const int*)  d_in[2];
    const float* eps0  = (const float*)d_in[3];
    const float* w1_0  = (const float*)d_in[4];
    const float* b1_0  = (const float*)d_in[5];
    const float* w2_0  = (const float*)d_in[6];
    const float* b2_0  = (const float*)d_in[7];
    const float* eps_r = (const float*)d_in[8];
    const float* w1_r  = (const float*)d_in[9];
    const float* b1_r  = (const float*)d_in[10];
    const float* w2_r  = (const float*)d_in[11];
    const float* b2_r  = (const float*)d_in[12];
    const float* w_out = (const float*)d_in[13];
    const float* b_out = (const float*)d_in[14];
    float* pred = (float*)d_out;

    // workspace layout
    float* agg    = (float*)d_ws;                          // N*64
    float* hA     = agg + (size_t)N_NODES * HID;           // N*64
    float* hB     = hA  + (size_t)N_NODES * HID;           // N*64
    float* pooled = hB  + (size_t)N_NODES * HID;           // G*64

    const int* src = eidx;
    const int* dst = eidx + N_EDGES;

    const int num_tiles = (N_NODES + 15) / 16;             // 6250
    const int mlp_blocks = (num_tiles + TILES_PER_BLOCK - 1) / TILES_PER_BLOCK;
    const int agg_elems = N_NODES * HID;

    // ---- layer 0 (30 -> 64) ----
    zero_f32<<<(agg_elems + 255) / 256, 256, 0, stream>>>(agg, agg_elems);
    {
        long tot = (long)N_EDGES << 5;                     // 32 lanes/edge, C=30
        scatter_edges<<<(int)((tot + 255) / 256), 256, 0, stream>>>(
            x, IN_CH, IN_CH, 5, src, dst, agg, tot);
    }
    gin_mlp<<<mlp_blocks, 128, 0, stream>>>(
        x, IN_CH, IN_CH, 32, 5, agg, eps0, 0, w1_0, b1_0, w2_0, b2_0, hA, num_tiles);

    // ---- layers 1..3 (64 -> 64) ----
    float* hin = hA; float* ho = hB;
    for (int l = 0; l < 3; ++l) {
        zero_f32<<<(agg_elems + 255) / 256, 256, 0, stream>>>(agg, agg_elems);
        long tot = (long)N_EDGES << 6;
        scatter_edges<<<(int)((tot + 255) / 256), 256, 0, stream>>>(
            hin, HID, HID, 6, src, dst, agg, tot);
        gin_mlp<<<mlp_blocks, 128, 0, stream>>>(
            hin, HID, HID, HID, 6, agg,
            eps_r, l,
            w1_r + (size_t)l * HID * HID, b1_r + (size_t)l * HID,
            w2_r + (size_t)l * HID * HID, b2_r + (size_t)l * HID,
            ho, num_tiles);
        float* t = hin; hin = ho; ho = t;
    }

    // ---- pool + readout ----
    const int pool_elems = N_GRAPHS * HID;
    zero_f32<<<(pool_elems + 255) / 256, 256, 0, stream>>>(pooled, pool_elems);
    {
        long tot = (long)N_NODES * HID;
        pool_nodes<<<(int)((tot + 255) / 256), 256, 0, stream>>>(hin, batch, pooled, tot);
    }
    predict<<<N_GRAPHS, HID, 0, stream>>>(pooled, w_out, b_out, pred);
}